// LSTMP_39728447488304
// MI455X (gfx1250) — compile-verified
//
#include <hip/hip_runtime.h>
#include <stdint.h>

// ---------------- problem constants ----------------
#define T_   512
#define N_   64
#define C_   512
#define H_   1024
#define P_   512
#define G4_  4096            // 4*H
#define NBLK 32
#define NTHR 256

typedef __attribute__((ext_vector_type(2))) float v2f;
typedef __attribute__((ext_vector_type(4))) float v4f;
typedef __attribute__((ext_vector_type(8))) float v8f;
typedef __attribute__((ext_vector_type(4))) int   v4i_;

typedef __attribute__((address_space(1))) v4i_* g128_t;   // global int4*
typedef __attribute__((address_space(3))) v4i_* l128_t;   // LDS int4*

__device__ __forceinline__ v8f wmma_f32(v2f a, v2f b, v8f c) {
  // D = A(16x4 f32) * B(4x16 f32) + C(16x16 f32)
  return __builtin_amdgcn_wmma_f32_16x16x4_f32(
      false, a, false, b, (short)0, c, false, false);
}

__device__ __forceinline__ float sigmoidf_(float x) {
  return 1.0f / (1.0f + __expf(-x));
}

// XOR-swizzled index into a 16 x 1024 f32 panel (64 KB).
// Granule (4 floats) XOR-rotated by 4*row inside each 64-float block:
// conflict-free for b64 fragment reads and b128 staging stores,
// and the XOR folds to one v_xor_b32 against the unrolled constant k.
__device__ __forceinline__ int lidxx(int m, int k) {
  return (m << 10) + (k ^ (m << 2));
}

// global -> LDS 16-byte copy via ASYNCcnt-tracked DMA.
__device__ __forceinline__ void lds_cp4(float* lp, const float* gp) {
#if __has_builtin(__builtin_amdgcn_global_load_async_to_lds_b128)
  __builtin_amdgcn_global_load_async_to_lds_b128(
      (g128_t)gp, (l128_t)lp, 0, 0);
#else
  *(v4f*)lp = *(const v4f*)gp;
#endif
}

__device__ __forceinline__ void wait_async0() {
#if __has_builtin(__builtin_amdgcn_s_wait_asynccnt)
  __builtin_amdgcn_s_wait_asynccnt(0);
#else
  asm volatile("s_wait_asynccnt 0x0" ::: "memory");
#endif
}

// device-wide sense-reversing barrier (all NBLK blocks resident)
__device__ __forceinline__ void grid_barrier(unsigned* cnt, unsigned* gen) {
  __syncthreads();
  if (threadIdx.x == 0) {
    __threadfence();
    unsigned g = __hip_atomic_load(gen, __ATOMIC_RELAXED, __HIP_MEMORY_SCOPE_AGENT);
    unsigned a = __hip_atomic_fetch_add(cnt, 1u, __ATOMIC_ACQ_REL, __HIP_MEMORY_SCOPE_AGENT);
    if (a == NBLK - 1u) {
      __hip_atomic_store(cnt, 0u, __ATOMIC_RELAXED, __HIP_MEMORY_SCOPE_AGENT);
      __hip_atomic_fetch_add(gen, 1u, __ATOMIC_RELEASE, __HIP_MEMORY_SCOPE_AGENT);
    } else {
      while (__hip_atomic_load(gen, __ATOMIC_ACQUIRE, __HIP_MEMORY_SCOPE_AGENT) == g) {
        __builtin_amdgcn_s_sleep(2);
      }
    }
  }
  __syncthreads();
}

__global__ __launch_bounds__(NTHR)
void lstmp_persistent(const float* __restrict__ x,        // (T,N,C)
                      const int*   __restrict__ lengths,  // (N)
                      const float* __restrict__ wx_w,     // (4H,C)
                      const float* __restrict__ wx_b,     // (4H)
                      const float* __restrict__ wr_w,     // (4H,P)
                      const float* __restrict__ wo_w,     // (P,H)
                      const float* __restrict__ wc,       // (3H)
                      float*       __restrict__ out,      // (T,N,P)
                      unsigned*    __restrict__ bar,      // [0]=cnt [1]=gen
                      float*       __restrict__ g_buf,    // (N,4H)
                      float*       __restrict__ c_state,  // (N,H)
                      float*       __restrict__ a_buf,    // (N,H)
                      float*       __restrict__ r_buf)    // (N,P)
{
  __shared__ float ldsA[2][16 * 1024];    // 128 KB: two 16-row K=1024 panels

  const int tid   = threadIdx.x;
  const int lane  = tid & 31;
  const int wv    = tid >> 5;             // 0..7
  const int m     = lane & 15;            // A row / B col / D col in tile
  const int kb    = (lane >> 4) << 1;     // K sub-offset 0 or 2
  const int rhalf = (lane >> 4) << 3;     // D row half 0 or 8

  const int n0  = (blockIdx.x & 1) << 5;  // 32-row slab owned by this block
  const int grp = blockIdx.x >> 1;        // 0..15: 256-column group

  for (int t = 0; t < T_; ++t) {
    // ---- stage A-slab = [x_t | r_prev] rows n0..n0+31 into LDS ----
    {
      const float* xbase = x + ((size_t)t * N_ + n0) * C_;
      for (int idx = tid; idx < 32 * 256; idx += NTHR) {
        const int row = idx >> 8;                    // 0..31
        const int k4  = (idx & 255) << 2;            // 0..1020
        const float* src = (k4 < C_)
            ? (xbase + (size_t)row * C_ + k4)
            : (r_buf + (size_t)(n0 + row) * P_ + (k4 - C_));
        lds_cp4(&ldsA[row >> 4][lidxx(row & 15, k4)], src);
      }
      wait_async0();
      __syncthreads();
    }

    // ---- Phase 1: g[n0:n0+32, grp*256 : +256]; 2x2 tiles per wave ----
    {
      const int j0 = (grp << 8) + (wv << 5);
      v8f acc00 = {}, acc01 = {}, acc10 = {}, acc11 = {};
      const float* w0x = wx_w + (size_t)(j0 + m) * C_;
      const float* w1x = wx_w + (size_t)(j0 + 16 + m) * C_;
      #pragma unroll 4
      for (int k = 0; k < C_; k += 4) {
        v2f a0 = *(const v2f*)&ldsA[0][lidxx(m, k) + kb];
        v2f a1 = *(const v2f*)&ldsA[1][lidxx(m, k) + kb];
        v2f b0 = { w0x[k + kb], w0x[k + kb + 1] };
        v2f b1 = { w1x[k + kb], w1x[k + kb + 1] };
        acc00 = wmma_f32(a0, b0, acc00);
        acc01 = wmma_f32(a0, b1, acc01);
        acc10 = wmma_f32(a1, b0, acc10);
        acc11 = wmma_f32(a1, b1, acc11);
      }
      const float* w0r = wr_w + (size_t)(j0 + m) * P_;
      const float* w1r = wr_w + (size_t)(j0 + 16 + m) * P_;
      #pragma unroll 4
      for (int k = 0; k < P_; k += 4) {
        v2f a0 = *(const v2f*)&ldsA[0][lidxx(m, C_ + k) + kb];
        v2f a1 = *(const v2f*)&ldsA[1][lidxx(m, C_ + k) + kb];
        v2f b0 = { w0r[k + kb], w0r[k + kb + 1] };
        v2f b1 = { w1r[k + kb], w1r[k + kb + 1] };
        acc00 = wmma_f32(a0, b0, acc00);
        acc01 = wmma_f32(a0, b1, acc01);
        acc10 = wmma_f32(a1, b0, acc10);
        acc11 = wmma_f32(a1, b1, acc11);
      }
      const float bias0 = wx_b[j0 + m];
      const float bias1 = wx_b[j0 + 16 + m];
      #pragma unroll
      for (int v = 0; v < 8; ++v) {
        float* gr0 = g_buf + (size_t)(n0 + rhalf + v) * G4_;
        float* gr1 = g_buf + (size_t)(n0 + 16 + rhalf + v) * G4_;
        gr0[j0 + m]      = acc00[v] + bias0;
        gr0[j0 + 16 + m] = acc01[v] + bias1;
        gr1[j0 + m]      = acc10[v] + bias0;
        gr1[j0 + 16 + m] = acc11[v] + bias1;
      }
    }
    grid_barrier(bar, bar + 1);

    // ---- Phase 2: gates, peephole, cell state (float4 vectorized) ----
    for (int idx = blockIdx.x * NTHR + tid; idx < (N_ * H_) / 4;
         idx += NBLK * NTHR) {
      const int n  = idx >> 8;                 // H/4 = 256 vectors per n
      const int h4 = (idx & 255) << 2;
      const float* gr = g_buf + (size_t)n * G4_;
      const v4f i_o = *(const v4f*)&gr[h4];
      const v4f f_o = *(const v4f*)&gr[H_ + h4];
      const v4f o_o = *(const v4f*)&gr[2 * H_ + h4];
      const v4f c_o = *(const v4f*)&gr[3 * H_ + h4];
      const v4f wci = *(const v4f*)&wc[h4];
      const v4f wcf = *(const v4f*)&wc[H_ + h4];
      const v4f wco = *(const v4f*)&wc[2 * H_ + h4];
      const v4f cp  = *(const v4f*)&c_state[(size_t)n * H_ + h4];
      v4f cc, aa;
      #pragma unroll
      for (int q = 0; q < 4; ++q) {
        const float ig = sigmoidf_(i_o[q] + wci[q] * cp[q]);
        const float fg = sigmoidf_(f_o[q] + wcf[q] * cp[q]);
        const float c2 = fg * cp[q] + ig * tanhf(c_o[q]);
        const float og = sigmoidf_(o_o[q] + wco[q] * c2);
        cc[q] = c2;
        aa[q] = og * tanhf(c2);
      }
      *(v4f*)&c_state[(size_t)n * H_ + h4] = cc;
      *(v4f*)&a_buf[(size_t)n * H_ + h4]   = aa;
    }
    grid_barrier(bar, bar + 1);

    // ---- Phase 3: r = a @ wo_w^T ; masked store to out[t] ----
    {
      // restage both panels with a_buf rows n0..n0+31 (K = H = 1024)
      const float* abase = a_buf + (size_t)n0 * H_;
      for (int idx = tid; idx < 32 * 256; idx += NTHR) {
        const int row = idx >> 8;
        const int k4  = (idx & 255) << 2;
        lds_cp4(&ldsA[row >> 4][lidxx(row & 15, k4)],
                abase + (size_t)row * H_ + k4);
      }
      wait_async0();
      __syncthreads();

      if (wv < 2) {                        // 2 p-tiles x 2 n-subtiles / block
        const int p0 = (grp << 5) + (wv << 4);
        v8f acc0 = {}, acc1 = {};
        const float* w0 = wo_w + (size_t)(p0 + m) * H_;
        #pragma unroll 4
        for (int k = 0; k < H_; k += 4) {
          v2f a0 = *(const v2f*)&ldsA[0][lidxx(m, k) + kb];
          v2f a1 = *(const v2f*)&ldsA[1][lidxx(m, k) + kb];
          v2f b  = { w0[k + kb], w0[k + kb + 1] };
          acc0 = wmma_f32(a0, b, acc0);
          acc1 = wmma_f32(a1, b, acc1);
        }
        #pragma unroll
        for (int v = 0; v < 8; ++v) {
          const int na = n0 + rhalf + v;
          const int nb2 = n0 + 16 + rhalf + v;
          const float v0 = acc0[v], v1 = acc1[v];
          r_buf[(size_t)na * P_ + (p0 + m)]  = v0;
          r_buf[(size_t)nb2 * P_ + (p0 + m)] = v1;
          out[((size_t)t * N_ + na) * P_ + (p0 + m)] =
              (t < lengths[na]) ? v0 : 0.0f;
          out[((size_t)t * N_ + nb2) * P_ + (p0 + m)] =
              (t < lengths[nb2]) ? v1 : 0.0f;
        }
      }
    }
    grid_barrier(bar, bar + 1);
  }
}

extern "C" void kernel_launch(void* const* d_in, const int* in_sizes, int n_in,
                              void* d_out, int out_size, void* d_ws, size_t ws_size,
                              hipStream_t stream) {
  const float* x    = (const float*)d_in[0];
  const int*   len  = (const int*)d_in[1];     // jax x64 off -> int32
  const float* wx_w = (const float*)d_in[2];
  const float* wx_b = (const float*)d_in[3];
  const float* wr_w = (const float*)d_in[4];
  const float* wo_w = (const float*)d_in[5];
  const float* wc   = (const float*)d_in[6];
  float* out = (float*)d_out;

  char* ws = (char*)d_ws;
  unsigned* bar  = (unsigned*)ws;                                   // 256 B
  float* g_buf   = (float*)(ws + 256);                              // 1 MB
  float* c_state = g_buf + (size_t)N_ * G4_;                        // 256 KB
  float* a_buf   = c_state + (size_t)N_ * H_;                       // 256 KB
  float* r_buf   = a_buf + (size_t)N_ * H_;                         // 128 KB

  (void)hipMemsetAsync(bar, 0, 256, stream);
  (void)hipMemsetAsync(c_state, 0, (size_t)N_ * H_ * sizeof(float), stream);
  (void)hipMemsetAsync(r_buf, 0, (size_t)N_ * P_ * sizeof(float), stream);

  lstmp_persistent<<<NBLK, NTHR, 0, stream>>>(
      x, len, wx_w, wx_b, wr_w, wo_w, wc, out,
      bar, g_buf, c_state, a_buf, r_buf);
}